// UniformMatcher_79877801771078
// MI455X (gfx1250) — compile-verified
//
#include <hip/hip_runtime.h>
#include <hip/hip_bf16.h>
#include <stdint.h>

// Problem constants (fixed by the reference's setup_inputs)
#define BS      8
#define NQ      16384
#define T_TOT   512          // BS * NT = 8 * 64 targets (full set used per image: source bug)
#define MT      4            // match_times
#define TPB     512          // 16 waves of 32
#define TTILE   16           // target rows per block (one per wave)
#define CHUNK_Q 2048         // queries staged per LDS chunk (32 KB of float4)
#define NCHUNK  (NQ / CHUNK_Q)

// Lexicographic "ranks higher in top-k" comparator matching jax.lax.top_k on
// negated cost: larger value first; ties -> smaller index first.
__device__ __forceinline__ bool better(float av, int ai, float bv, int bi) {
  return (av > bv) || ((av == bv) && (ai < bi));
}

// Branchless compare-exchange: after the call, (av,ai) ranks >= (bv,bi).
// Lowers to v_cmp + v_cndmask chains — no divergent control flow.
__device__ __forceinline__ void cswap(float& av, int& ai, float& bv, int& bi) {
  const bool sw = better(bv, bi, av, ai);
  const float hv = sw ? bv : av;
  const float lv = sw ? av : bv;
  const int   hi = sw ? bi : ai;
  const int   li = sw ? ai : bi;
  av = hv; bv = lv; ai = hi; bi = li;
}

// Hot-path insert: single-v_cmp pre-filter (superset: includes value ties),
// then a fully branchless tie-aware conditional insert + bubble.
__device__ __forceinline__ void insert4_hot(float v, int i, float tv[4], int ti[4]) {
  if (v >= tv[3]) {                      // one compare; rarely true in steady state
    const bool ok = better(v, i, tv[3], ti[3]);
    tv[3] = ok ? v : tv[3];
    ti[3] = ok ? i : ti[3];
    cswap(tv[2], ti[2], tv[3], ti[3]);
    cswap(tv[1], ti[1], tv[2], ti[2]);
    cswap(tv[0], ti[0], tv[1], ti[1]);
  }
}

// Cold-path insert (butterfly merge epilogue): same semantics.
__device__ __forceinline__ void insert4(float v, int i, float tv[4], int ti[4]) {
  if (better(v, i, tv[3], ti[3])) {
    tv[3] = v; ti[3] = i;
    cswap(tv[2], ti[2], tv[3], ti[3]);
    cswap(tv[1], ti[1], tv[2], ti[2]);
    cswap(tv[0], ti[0], tv[1], ti[1]);
  }
}

// Stage one 2048-query chunk (32 KB) into LDS with CDNA5 async global->LDS DMA.
// Each of the 512 threads issues 4 coalesced b128 transfers (ASYNCcnt-tracked).
__device__ __forceinline__ void stage_chunk(const float4* __restrict__ gbase,
                                            float4* lbuf, int tid) {
#pragma unroll
  for (int i = 0; i < 4; ++i) {
    const int e = tid + i * TPB;
    // Low 32 bits of a generic shared pointer == workgroup-relative LDS byte addr.
    uint32_t lds_addr = (uint32_t)(uintptr_t)(lbuf + e);
    const float4* g = gbase + e;
    asm volatile("global_load_async_to_lds_b128 %0, %1, off"
                 :: "v"(lds_addr), "v"(g)
                 : "memory");
  }
}

__global__ __launch_bounds__(TPB)
void UniformMatcher_cdna5_kernel(const float* __restrict__ pred_boxes,
                                 const float* __restrict__ targets,
                                 int* __restrict__ out) {
  __shared__ float4 buf[2][CHUNK_Q];   // 64 KB double buffer (of 320 KB/WGP)

  const int tid  = threadIdx.x;
  const int wave = tid >> 5;
  const int lane = tid & 31;
  const int b     = blockIdx.x >> 5;   // 32 target-tiles per batch
  const int ttile = blockIdx.x & 31;
  const int t     = ttile * TTILE + wave;   // this wave's global target row

  const float4* gpred = (const float4*)pred_boxes + (size_t)b * NQ;
  const float4  tg    = ((const float4*)targets)[t];

  // Running top-4 of (-cost, query index), sorted desc.
  float tv[4] = {-1e30f, -1e30f, -1e30f, -1e30f};
  int   ti[4] = {0x7fffffff, 0x7fffffff, 0x7fffffff, 0x7fffffff};

  // Prologue: stage chunk 0.
  stage_chunk(gpred, &buf[0][0], tid);

  for (int c = 0; c < NCHUNK; ++c) {
    if (c + 1 < NCHUNK) {
      // Prefetch next chunk into the other buffer, then wait only for the
      // current chunk (async loads complete in order -> ASYNCcnt <= 4).
      stage_chunk(gpred + (size_t)(c + 1) * CHUNK_Q, &buf[(c + 1) & 1][0], tid);
      asm volatile("s_wait_asynccnt 4" ::: "memory");
    } else {
      asm volatile("s_wait_asynccnt 0" ::: "memory");
    }
    __syncthreads();   // all threads' async writes for chunk c visible in LDS

    const float4* lb = &buf[c & 1][0];
    const int qbase = c * CHUNK_Q;
#pragma unroll 8
    for (int it = 0; it < CHUNK_Q / 32; ++it) {
      const int ql = lane + it * 32;          // conflict-free b128 LDS reads
      const float4 p = lb[ql];
      // abs() folds into VALU source modifiers of the adds.
      const float cost = fabsf(tg.x - p.x) + fabsf(tg.y - p.y) +
                         fabsf(tg.z - p.z) + fabsf(tg.w - p.w);
      insert4_hot(-cost, qbase + ql, tv, ti);
    }
    __syncthreads();   // done reading this buffer before it is overwritten
  }

  // Wave32 butterfly merge of per-lane top-4 lists (log2(32) = 5 steps).
#pragma unroll
  for (int mask = 16; mask >= 1; mask >>= 1) {
    float ov[4]; int oi[4];
#pragma unroll
    for (int k = 0; k < 4; ++k) {
      ov[k] = __shfl_xor(tv[k], mask, 32);
      oi[k] = __shfl_xor(ti[k], mask, 32);
    }
#pragma unroll
    for (int k = 0; k < 4; ++k) insert4(ov[k], oi[k], tv, ti);
  }

  if (lane == 0) {
    // idx_i: [BS, T*2*MT] row-major, then idx_j of same shape appended.
    const size_t row = (size_t)b * (T_TOT * 2 * MT) + (size_t)t * (2 * MT);
    int* idx_i = out + row;
    int* idx_j = out + (size_t)BS * T_TOT * 2 * MT + row;
#pragma unroll
    for (int k = 0; k < MT; ++k) {
      idx_i[k]      = ti[k];   // pred top-k
      idx_i[MT + k] = ti[k];   // anchor top-k == pred top-k (source bug)
      idx_j[k]      = k;
      idx_j[MT + k] = k;
    }
  }
}

extern "C" void kernel_launch(void* const* d_in, const int* in_sizes, int n_in,
                              void* d_out, int out_size, void* d_ws, size_t ws_size,
                              hipStream_t stream) {
  const float* pred_boxes = (const float*)d_in[0];
  // d_in[1] (anchors) is dead: reference sets C_anchors = C_pred.
  const float* targets    = (const float*)d_in[2];
  // d_in[3] (match_times) == 4, baked into MT.
  (void)in_sizes; (void)n_in; (void)d_ws; (void)ws_size; (void)out_size;

  int* out = (int*)d_out;
  dim3 grid(BS * (T_TOT / TTILE));   // 8 * 32 = 256 blocks
  dim3 block(TPB);                   // 16 waves each
  hipLaunchKernelGGL(UniformMatcher_cdna5_kernel, grid, block, 0, stream,
                     pred_boxes, targets, out);
}